// BatchDotPred_27092653703581
// MI455X (gfx1250) — compile-verified
//
#include <hip/hip_runtime.h>

typedef __attribute__((ext_vector_type(2))) float v2f;
typedef __attribute__((ext_vector_type(8))) float v8f;

#define D_FEAT 128
#define EDGES_PER_WAVE 16
#define WAVES_PER_BLOCK 8
#define EDGES_PER_BLOCK (EDGES_PER_WAVE * WAVES_PER_BLOCK)

// One wave computes 16 edge scores:
//   A (16x4)  = src feature rows, M = edge-in-tile, K = feature chunk
//   B (4x16)  = dst feature rows, N = edge-in-tile, K = feature chunk
//   D = A x B + C accumulated over K=128 in steps of 4 (fp32 WMMA)
//   scores = diag(D)
__global__ __launch_bounds__(256) void BatchDotPred_wmma_kernel(
    const int* __restrict__ edges,   // [E][2] int32
    const float* __restrict__ feat,  // [N][128] fp32
    float* __restrict__ out,         // [E] fp32
    int n_edges)
{
    const int lane        = threadIdx.x & 31;
    const int waveInBlock = threadIdx.x >> 5;
    const long long edgeBase =
        (long long)blockIdx.x * EDGES_PER_BLOCK + (long long)waveInBlock * EDGES_PER_WAVE;
    if (edgeBase >= n_edges) return;            // wave-uniform: EXEC stays all-1s

    const int m  = lane & 15;                   // edge-within-tile (M for A, N for B)
    const int hi = lane >> 4;                   // half-wave selects K pair {0,1} vs {2,3}

    long long e = edgeBase + m;
    if (e >= n_edges) e = (long long)n_edges - 1;   // clamp tail; keeps all lanes live
    const int src = edges[2 * e + 0];
    const int dst = edges[2 * e + 1];

    const float* __restrict__ srow = feat + (long long)src * D_FEAT;
    const float* __restrict__ drow = feat + (long long)dst * D_FEAT;

    v8f c0 = {0.f, 0.f, 0.f, 0.f, 0.f, 0.f, 0.f, 0.f};
    v8f c1 = {0.f, 0.f, 0.f, 0.f, 0.f, 0.f, 0.f, 0.f};

    // f32 WMMA operand layout (2 VGPRs per matrix):
    //   A vgpr0 = A[m][k0+2*hi], vgpr1 = A[m][k0+2*hi+1]   (lanes 0-15: K=k0,k0+1; 16-31: K=k0+2,k0+3)
    //   B vgpr0 = B[k0+2*hi][n], vgpr1 = B[k0+2*hi+1][n]   with B[k][n] = feat[dst[n]][k]
    // Each lane loads one consecutive float2 per matrix per chunk -> global_load_b64.
#pragma unroll
    for (int k0 = 0; k0 < D_FEAT; k0 += 8) {
        const int ka = k0 + 2 * hi;
        v2f a0 = { srow[ka],     srow[ka + 1] };
        v2f b0 = { drow[ka],     drow[ka + 1] };
        v2f a1 = { srow[ka + 4], srow[ka + 5] };
        v2f b1 = { drow[ka + 4], drow[ka + 5] };
        c0 = __builtin_amdgcn_wmma_f32_16x16x4_f32(false, a0, false, b0,
                                                   (short)0, c0, false, false);
        c1 = __builtin_amdgcn_wmma_f32_16x16x4_f32(false, a1, false, b1,
                                                   (short)0, c1, false, false);
    }

    v8f c = c0 + c1;

    // Diagonal extraction. C/D layout: VGPR v holds (M=v, N=lane) for lanes 0-15
    // and (M=v+8, N=lane-16) for lanes 16-31. So diag element m=v sits in c[v] at
    // lane v, and diag element m=v+8 sits in c[v] at lane v+24.
    float my = 0.f;
#pragma unroll
    for (int v = 0; v < 8; ++v) {
        float dlo = __shfl(c[v], v, 32);        // score for edge-in-tile v
        float dhi = __shfl(c[v], v + 24, 32);   // score for edge-in-tile v+8
        my = (lane == v)     ? dlo : my;
        my = (lane == v + 8) ? dhi : my;
    }

    const long long eo = edgeBase + lane;
    if (lane < EDGES_PER_WAVE && eo < n_edges) {
        out[eo] = my;
    }
}

extern "C" void kernel_launch(void* const* d_in, const int* in_sizes, int n_in,
                              void* d_out, int out_size, void* d_ws, size_t ws_size,
                              hipStream_t stream) {
    const int*   edges = (const int*)d_in[0];    // [E][2] int32
    const float* feat  = (const float*)d_in[1];  // [N][128] fp32
    float*       out   = (float*)d_out;          // [E] fp32

    const int n_edges = in_sizes[0] / 2;
    const int blocks  = (n_edges + EDGES_PER_BLOCK - 1) / EDGES_PER_BLOCK;

    BatchDotPred_wmma_kernel<<<blocks, 256, 0, stream>>>(edges, feat, out, n_edges);
}